// ThaiW2PTFModel_34368328302849
// MI455X (gfx1250) — compile-verified
//
#include <hip/hip_runtime.h>
#include <hip/hip_bf16.h>
#include <math.h>

// Model dims (compile-time constants from the reference)
#define Gv   100
#define Pv   120
#define Ev   128
#define Hv   128
#define Tv   2048
#define TH3  384     // 3*H
#define MAXD 20
#define SOS_ID 2
#define EOS_ID 3

typedef float v2f __attribute__((ext_vector_type(2)));
typedef float v4f __attribute__((ext_vector_type(4)));
typedef float v8f __attribute__((ext_vector_type(8)));

// ---------------------------------------------------------------------------
// Kernel A: XW[t][j] = sum_e enc_emb[x_ids[t]][e] * enc_w_ih[j][e] + enc_b_ih[j]
// (2048 x 128) @ (128 x 384) fp32 GEMM via V_WMMA_F32_16X16X4_F32.
// One wave per 16x16 output tile; 32 k-steps of K=4.
// ---------------------------------------------------------------------------
__global__ __launch_bounds__(256) void enc_xproj_wmma(
    const int*   __restrict__ x_ids,   // [T]
    const float* __restrict__ emb,     // [G][E]
    const float* __restrict__ w_ih,    // [3H][E]
    const float* __restrict__ b_ih,    // [3H]
    float*       __restrict__ xw)      // [T][3H]
{
    const int NT   = TH3 / 16;                 // 24 N-tiles
    const int wave = threadIdx.x >> 5;         // 8 waves/block
    const int lane = threadIdx.x & 31;
    const int tile = blockIdx.x * 8 + wave;    // 0 .. 3071
    const int mt   = tile / NT;
    const int nt   = tile - mt * NT;

    const int l    = lane & 15;
    const int hi   = lane >> 4;                // 0 = K{0,1}, 1 = K{2,3}
    const int koff = hi << 1;

    const int m = mt * 16 + l;                 // encoder time index for this lane
    const float* __restrict__ xrow = emb  + (size_t)x_ids[m] * Ev;
    const float* __restrict__ wrow = w_ih + (size_t)(nt * 16 + l) * Ev;

    v8f c = {0.f, 0.f, 0.f, 0.f, 0.f, 0.f, 0.f, 0.f};
#pragma unroll
    for (int k0 = 0; k0 < Ev; k0 += 4) {
        v2f a = *(const v2f*)(xrow + k0 + koff);   // 8B-aligned: k0%4==0
        v2f b = *(const v2f*)(wrow + k0 + koff);
        c = __builtin_amdgcn_wmma_f32_16x16x4_f32(
                /*neg_a=*/false, a, /*neg_b=*/false, b,
                /*c_mod=*/(short)0, c, /*reuse_a=*/false, /*reuse_b=*/false);
    }

    const int col  = nt * 16 + l;
    const float bias = b_ih[col];
#pragma unroll
    for (int v = 0; v < 8; ++v) {
        const int row = mt * 16 + v + hi * 8;
        xw[(size_t)row * TH3 + col] = c[v] + bias;
    }
}

// ---------------------------------------------------------------------------
// Kernel B: persistent single-workgroup sequential GRU (encoder 2048 steps)
// + 20-step greedy decoder. 384 threads = 12 wave32s on one WGP.
//
// Key change vs v1: each thread pins its 128-float row of enc_w_hh in VGPRs
// (32 x v4f = 128 registers) for the whole encoder loop. Per step the only
// memory traffic is the 128-float h vector (LDS broadcast) and one 1.5 KB
// xw row (hoisted global load, overlapped with the matvec).
// ---------------------------------------------------------------------------
__device__ __forceinline__ float sigmoidf_(float x) {
    return 1.0f / (1.0f + __expf(-x));
}
__device__ __forceinline__ float tanhf_(float x) {
    // tanh(x) = 1 - 2/(exp(2x)+1)  (fast hw exp path, avoids libcall)
    return 1.0f - 2.0f / (__expf(2.0f * x) + 1.0f);
}

__global__ __launch_bounds__(TH3) void seq_gru_decode(
    const float* __restrict__ xw,       // [T][3H] precomputed ih preact (+b_ih)
    const float* __restrict__ enc_w_hh, // [3H][H]
    const float* __restrict__ enc_b_hh, // [3H]
    const float* __restrict__ dec_emb,  // [P][E]
    const float* __restrict__ dec_w_ih, // [3H][E]
    const float* __restrict__ dec_w_hh, // [3H][H]
    const float* __restrict__ dec_b_ih, // [3H]
    const float* __restrict__ dec_b_hh, // [3H]
    const float* __restrict__ fc_w,     // [P][H]
    const float* __restrict__ fc_b,     // [P]
    float*       __restrict__ out)      // [MAXD preds][MAXD*P logits]
{
    __shared__ float hbuf[2][Hv];
    __shared__ float aih[TH3];
    __shared__ float ahh[TH3];
    __shared__ float decv[Ev];
    __shared__ float hn[Hv];
    __shared__ float logits[Pv];
    __shared__ int   s_pred;
    __shared__ int   s_accept;
    __shared__ int   s_ended;

    const int tid = threadIdx.x;

    // Pin this thread's recurrent weight row in registers for the whole loop.
    v4f wreg[Hv / 4];
    {
        const float* __restrict__ wr = enc_w_hh + (size_t)tid * Hv;
#pragma unroll
        for (int i = 0; i < Hv / 4; ++i) wreg[i] = *(const v4f*)(wr + 4 * i);
    }
    const float bhh = enc_b_hh[tid];

    if (tid < Hv) { hbuf[0][tid] = 0.f; hbuf[1][tid] = 0.f; }
    if (tid == 0) { s_ended = 0; }
    __syncthreads();

    // ---------------- Encoder: 2048 sequential steps ----------------
    for (int t = 0; t < Tv; ++t) {
        const float* __restrict__ h  = hbuf[t & 1];
        float*       __restrict__ h2 = hbuf[(t + 1) & 1];

        // Hoist the xw-row loads so global latency overlaps the matvec.
        float x0 = 0.f, x1 = 0.f, x2 = 0.f;
        if (tid < Hv) {
            const float* __restrict__ xr = xw + (size_t)t * TH3;
            x0 = xr[tid];
            x1 = xr[Hv + tid];
            x2 = xr[2 * Hv + tid];
        }

        // rzn_hh[tid] = b_hh[tid] + h . W_hh[tid][:]
        // W row in VGPRs; h via conflict-free LDS broadcast (v4 loads).
        {
            float a0 = bhh, a1 = 0.f, a2 = 0.f, a3 = 0.f;
#pragma unroll
            for (int i = 0; i < Hv / 4; ++i) {
                const v4f hv = *(const v4f*)(h + 4 * i);
                a0 += hv.x * wreg[i].x;
                a1 += hv.y * wreg[i].y;
                a2 += hv.z * wreg[i].z;
                a3 += hv.w * wreg[i].w;
            }
            ahh[tid] = (a0 + a1) + (a2 + a3);
        }
        __syncthreads();

        if (tid < Hv) {
            const float r = sigmoidf_(x0 + ahh[tid]);
            const float z = sigmoidf_(x1 + ahh[Hv + tid]);
            const float n = tanhf_(   x2 + r * ahh[2 * Hv + tid]);
            h2[tid] = (1.f - z) * n + z * h[tid];
        }
        __syncthreads();
    }

    float* __restrict__ h = hbuf[Tv & 1];   // final encoder hidden state

    // ---------------- Decoder: 20 greedy steps ----------------
    if (tid < Ev) decv[tid] = dec_emb[SOS_ID * Ev + tid];
    __syncthreads();

    for (int d = 0; d < MAXD; ++d) {
        // Both preactivation vectors (ih on current input, hh on h)
        {
            const float* __restrict__ wi = dec_w_ih + (size_t)tid * Ev;
            const float* __restrict__ wh = dec_w_hh + (size_t)tid * Hv;
            float ai = dec_b_ih[tid];
            float ah = dec_b_hh[tid];
#pragma unroll 8
            for (int k = 0; k < Hv; ++k) {
                ai += decv[k] * wi[k];
                ah += h[k]    * wh[k];
            }
            aih[tid] = ai;
            ahh[tid] = ah;
        }
        __syncthreads();

        if (tid < Hv) {
            const float r = sigmoidf_(aih[tid]          + ahh[tid]);
            const float z = sigmoidf_(aih[Hv + tid]     + ahh[Hv + tid]);
            const float n = tanhf_(   aih[2 * Hv + tid] + r * ahh[2 * Hv + tid]);
            hn[tid] = (1.f - z) * n + z * h[tid];
        }
        __syncthreads();

        // logits = hn @ fc_w^T + fc_b ; masked logits written using PRE-step ended
        if (tid < Pv) {
            const float* __restrict__ wr = fc_w + (size_t)tid * Hv;
            float acc = fc_b[tid];
#pragma unroll 8
            for (int k = 0; k < Hv; ++k) acc += hn[k] * wr[k];
            logits[tid] = acc;
            out[MAXD + (size_t)d * Pv + tid] = s_ended ? 0.f : acc;
        }
        __syncthreads();

        if (tid == 0) {
            int best = 0; float bv = logits[0];
            for (int p = 1; p < Pv; ++p) {           // first-max, like jnp.argmax
                const float v = logits[p];
                if (v > bv) { bv = v; best = p; }
            }
            const int ended_prev = s_ended;
            s_pred   = best;
            s_accept = !ended_prev;                  // h/dec update uses PRE-step ended
            out[d]   = (!ended_prev && best != EOS_ID) ? (float)best : -1.f;
            s_ended  = ended_prev || (best == EOS_ID);
        }
        __syncthreads();

        if (s_accept && tid < Hv) {
            h[tid]    = hn[tid];
            decv[tid] = dec_emb[(size_t)s_pred * Ev + tid];
        }
        __syncthreads();
    }
}

// ---------------------------------------------------------------------------
extern "C" void kernel_launch(void* const* d_in, const int* in_sizes, int n_in,
                              void* d_out, int out_size, void* d_ws, size_t ws_size,
                              hipStream_t stream) {
    (void)in_sizes; (void)n_in; (void)out_size; (void)ws_size;

    const int*   x_ids    = (const int*)  d_in[0];
    const float* enc_emb  = (const float*)d_in[1];
    const float* enc_w_ih = (const float*)d_in[2];
    const float* enc_w_hh = (const float*)d_in[3];
    const float* enc_b_ih = (const float*)d_in[4];
    const float* enc_b_hh = (const float*)d_in[5];
    const float* dec_emb  = (const float*)d_in[6];
    const float* dec_w_ih = (const float*)d_in[7];
    const float* dec_w_hh = (const float*)d_in[8];
    const float* dec_b_ih = (const float*)d_in[9];
    const float* dec_b_hh = (const float*)d_in[10];
    const float* fc_w     = (const float*)d_in[11];
    const float* fc_b     = (const float*)d_in[12];

    float* out = (float*)d_out;
    float* xw  = (float*)d_ws;   // T*3H floats = 3 MB scratch

    // Kernel A: parallel input projection, 3072 WMMA tiles (8 waves/block).
    enc_xproj_wmma<<<(Tv / 16) * (TH3 / 16) / 8, 256, 0, stream>>>(
        x_ids, enc_emb, enc_w_ih, enc_b_ih, xw);

    // Kernel B: persistent sequential GRU + greedy decoder (one workgroup).
    seq_gru_decode<<<1, TH3, 0, stream>>>(
        xw, enc_w_hh, enc_b_hh, dec_emb, dec_w_ih, dec_w_hh,
        dec_b_ih, dec_b_hh, fc_w, fc_b, out);
}